// _FusedDslashAdapter_62723702391313
// MI455X (gfx1250) — compile-verified
//
#include <hip/hip_runtime.h>
#include <cstdint>

// Wilson-like Dslash on 16^4 lattice, 12 spin-color, complex split re/im.
// Memory-bound (~1 FLOP/byte, 604 MB K-stream): async global->LDS double
// buffering streams K; the per-site complex 12x12 matvec runs on the matrix
// pipe as V_WMMA_F32_16X16X4_F32 with A = {p_re; p_im} rows and B = K^T from LDS.

#define L 16
#define NSITE 65536          // 16^4
#define NSC 12
#define MAT 144              // 12x12
#define LINE_SITES 16        // one x-line per block
#define THREADS 64           // 2 wave32; each wave owns 8 sites of the line
#define STAGE_FLOATS (LINE_SITES * MAT)   // 2304 floats per (re or im) stage
#define STAGE_BYTES (STAGE_FLOATS * 4)    // 9216 B
#define DIAG 4.1f

typedef float v2f __attribute__((ext_vector_type(2)));
typedef float v8f __attribute__((ext_vector_type(8)));

// Only meaningful on the device pass; host pass cannot see AMDGCN builtins.
#if defined(__HIP_DEVICE_COMPILE__) && !__has_builtin(__builtin_amdgcn_wmma_f32_16x16x4_f32)
#error "missing __builtin_amdgcn_wmma_f32_16x16x4_f32 on this toolchain"
#endif

// s_wait_asynccnt with a literal immediate
#define WAIT_ASYNC(n) asm volatile("s_wait_asynccnt " #n ::: "memory")

// One 16B async global->LDS transfer (GVS form: saddr base + 32-bit voffset).
__device__ __forceinline__ void async_cp_b128(uint32_t lds_addr, uint32_t voff,
                                              const float* sbase) {
  asm volatile("global_load_async_to_lds_b128 %0, %1, %2"
               :: "v"(lds_addr), "v"(voff), "s"(sbase)
               : "memory");
}

// Stage one (mu,dir) K-block for 16 sites: 9216 B re + 9216 B im (contiguous in LDS).
// 64 lanes x 16 B = 1024 B per sweep -> 9 sweeps each. 18 async ops / wave / stage.
__device__ __forceinline__ void issue_stage(const float* gre, const float* gim,
                                            uint32_t lre, uint32_t lim, uint32_t tid) {
  uint32_t off = tid * 16u;
#pragma unroll
  for (int it = 0; it < 9; ++it) {
    async_cp_b128(lre + off, off, gre);
    async_cp_b128(lim + off, off, gim);
    off += THREADS * 16u;
  }
}

__device__ __forceinline__ float xorsign(float v, uint32_t m) {
  return __uint_as_float(__float_as_uint(v) ^ m);
}

__global__ __launch_bounds__(THREADS) void dslash_wmma(
    const float* __restrict__ psi_re, const float* __restrict__ psi_im,
    const float* __restrict__ Kfr, const float* __restrict__ Kfi,
    const float* __restrict__ Kbr, const float* __restrict__ Kbi,
    float* __restrict__ out) {
  __shared__ __align__(16) float kbuf[2][2][STAGE_FLOATS];  // [buffer][re/im][16 sites x 144]

  const uint32_t tid  = threadIdx.x;   // 0..63
  const uint32_t lane = tid & 31u;     // lane within wave32
  const uint32_t wv   = tid >> 5;      // wave id: sites wv*8 .. wv*8+7
  const int line = blockIdx.x;         // (t,z,y) line, x = 0..15
  const int base = line * LINE_SITES;
  const int y = line & 15;
  const int z = (line >> 4) & 15;
  const int t = (line >> 8) & 15;

  const uint32_t lmod  = lane & 15u;   // A row index m / B column index n
  const uint32_t lhalf = lane >> 4;    // k-half: lanes 16-31 carry k=2,3

  // LDS byte bases of the two double buffers (low 32 bits of flat shared ptr).
  uint32_t ldsbase[2];
  ldsbase[0] = (uint32_t)(uintptr_t)&kbuf[0][0][0];
  ldsbase[1] = (uint32_t)(uintptr_t)&kbuf[1][0][0];

  // Per-lane A-operand base pointers: A_re rows {0:p_re, 1:p_im}; A_im rows
  // {0:-p_im, 1:+p_re}. Lane parity selects the array; +2 floats for upper k-half.
  const float* baseA_re = ((lane & 1u) ? psi_im : psi_re) + lhalf * 2u;
  const float* baseA_im = ((lane & 1u) ? psi_re : psi_im) + lhalf * 2u;
  const uint32_t sgn = (lmod == 0u) ? 0x80000000u : 0u;  // negate row 0 of A_im

  // Per-stage K pointers (contiguous 16-site block per line).
  const float* kre_tab[8];
  const float* kim_tab[8];
#pragma unroll
  for (int st = 0; st < 8; ++st) {
    const int  mu  = st >> 1;
    const bool fwd = (st & 1) == 0;
    const size_t koff = ((size_t)mu * NSITE + (size_t)base) * MAT;
    kre_tab[st] = (fwd ? Kfr : Kbr) + koff;
    kim_tab[st] = (fwd ? Kfi : Kbi) + koff;
  }

  // One 16x16 f32 accumulator per site: row0 = c_re[n], row1 = c_im[n].
  v8f acc[8];
#pragma unroll
  for (int sl = 0; sl < 8; ++sl) acc[sl] = {};

  // Prologue: stage 0 into buffer 0.
  issue_stage(kre_tab[0], kim_tab[0], ldsbase[0], ldsbase[0] + STAGE_BYTES, tid);

#pragma unroll
  for (int st = 0; st < 8; ++st) {
    if (st < 7) {
      issue_stage(kre_tab[st + 1], kim_tab[st + 1],
                  ldsbase[(st + 1) & 1], ldsbase[(st + 1) & 1] + STAGE_BYTES, tid);
      WAIT_ASYNC(18);   // in-order: previous stage's 18 transfers have landed
    } else {
      WAIT_ASYNC(0);
    }
    __syncthreads();

    const int mu  = st >> 1;
    const int dlt = ((st & 1) == 0) ? 1 : 15;

#pragma unroll
    for (int sl = 0; sl < 8; ++sl) {
      const int xs = (int)(wv * 8u) + sl;          // this wave's site x-coord
      int cc[4] = { t, z, y, xs };
      cc[mu] = (cc[mu] + dlt) & 15;
      const int nbr = ((cc[0] * L + cc[1]) * L + cc[2]) * L + cc[3];

      // A operands: neighbor spinor chunks (L2-resident), lanes >=2 duplicate
      // lane 0/1 (finite garbage in unused D rows 2..15).
      const float* pa_re = baseA_re + (size_t)nbr * NSC;
      const float* pa_im = baseA_im + (size_t)nbr * NSC;

      // B operands: K^T chunks from LDS, one ds_load_b64 per chunk, bank-conflict-free.
      const int site = xs;
      const float* kb_re = &kbuf[st & 1][0][(size_t)site * MAT + lmod * NSC + lhalf * 2u];
      const float* kb_im = &kbuf[st & 1][1][(size_t)site * MAT + lmod * NSC + lhalf * 2u];

#pragma unroll
      for (int c = 0; c < 3; ++c) {
        v2f a_re = *(const v2f*)(pa_re + c * 4);
        v2f a_im = *(const v2f*)(pa_im + c * 4);
        a_im.x = xorsign(a_im.x, sgn);
        a_im.y = xorsign(a_im.y, sgn);
        v2f b_re = *(const v2f*)(kb_re + c * 4);
        v2f b_im = *(const v2f*)(kb_im + c * 4);
        // D[0][n] += p_re . Kre[n,:] ; D[1][n] += p_im . Kre[n,:]
        acc[sl] = __builtin_amdgcn_wmma_f32_16x16x4_f32(
            false, a_re, false, b_re, (short)0, acc[sl], false, false);
        // D[0][n] += -p_im . Kim[n,:] ; D[1][n] += p_re . Kim[n,:]
        acc[sl] = __builtin_amdgcn_wmma_f32_16x16x4_f32(
            false, a_im, false, b_im, (short)0, acc[sl], false, false);
      }
    }
    __syncthreads();  // done reading this buffer before it is overwritten
  }

  // Epilogue: lane n<12 holds c_re in acc[sl][0] (M=0,N=n) and c_im in acc[sl][1].
  if (lmod < 12u && lhalf == 0u) {
#pragma unroll
    for (int sl = 0; sl < 8; ++sl) {
      const int s = base + (int)(wv * 8u) + sl;
      const size_t o = (size_t)s * NSC + lmod;
      out[o] = DIAG * psi_re[o] - 0.5f * acc[sl][0];
      out[(size_t)NSITE * NSC + o] = DIAG * psi_im[o] - 0.5f * acc[sl][1];
    }
  }
}

extern "C" void kernel_launch(void* const* d_in, const int* in_sizes, int n_in,
                              void* d_out, int out_size, void* d_ws, size_t ws_size,
                              hipStream_t stream) {
  (void)in_sizes; (void)n_in; (void)out_size; (void)d_ws; (void)ws_size;
  const float* psi_re = (const float*)d_in[0];
  const float* psi_im = (const float*)d_in[1];
  const float* Kfr    = (const float*)d_in[2];
  const float* Kfi    = (const float*)d_in[3];
  const float* Kbr    = (const float*)d_in[4];
  const float* Kbi    = (const float*)d_in[5];
  float* out = (float*)d_out;

  dim3 grid(NSITE / LINE_SITES);  // 4096 blocks
  dim3 block(THREADS);            // 64 threads = 2 wave32
  hipLaunchKernelGGL(dslash_wmma, grid, block, 0, stream,
                     psi_re, psi_im, Kfr, Kfi, Kbr, Kbi, out);
}